// NormVoxels_67259187855688
// MI455X (gfx1250) — compile-verified
//
#include <hip/hip_runtime.h>
#include <hip/hip_bf16.h>

// Problem constants (match reference)
#define NV_T 8
#define NV_G 128
#define NV_RMIN (-1.5f)
#define NV_STEP (3.0f / 128.0f)

// ---------------------------------------------------------------------------
// Kernel 1: clone the 192 MiB voxel grid via the CDNA5 async global<->LDS
// data path. Branch-free: handles FULL tiles only (host guarantees this).
// The async ops add INST_OFFSET to BOTH the LDS and the global address
// so with the LDS slot stride equal to the
// global chunk stride (4096 B) all 16 chunks share one address VGPR pair and
// one LDS-address VGPR, driven purely by offset: immediates.
// 16 chunks/thread = 16-deep async queue per lane, 64 KB LDS/block.
// Side benefit vs SDMA memcpy: the stream pulls src AND dst through L2, so
// the scattered gathers/atomics of kernel 2 become L2 hits.
// ---------------------------------------------------------------------------
#define NV_CH 16  // float4 chunks per thread; tile = 256*NV_CH float4s

__global__ __launch_bounds__(256)
void nv_copy_async(const float4* __restrict__ src, float4* __restrict__ dst)
{
    __shared__ float4 buf[256 * NV_CH];
    const int tid = threadIdx.x;
    const size_t base = (size_t)blockIdx.x * (256 * NV_CH) + tid;
    // addrspace(3) byte offset of this thread's slot 0
    unsigned lds0 = (unsigned)(size_t)(&buf[tid]);

    const float4* s = src + base;
    float4*       d = dst + base;

#pragma unroll
    for (int k = 0; k < NV_CH; ++k) {
        asm volatile("global_load_async_to_lds_b128 %0, %1, off offset:%2"
                     :: "v"(lds0), "v"(s), "i"(k * 4096) : "memory");
    }
    asm volatile("s_wait_asynccnt 0" ::: "memory");

#pragma unroll
    for (int k = 0; k < NV_CH; ++k) {
        asm volatile("global_store_async_from_lds_b128 %0, %1, off offset:%2"
                     :: "v"(d), "v"(lds0), "i"(k * 4096) : "memory");
    }
    asm volatile("s_wait_asynccnt 0" ::: "memory");
}

// Scalar tail copy (only launched if total float4 count is not a multiple of
// the async tile; at reference sizes it never runs).
__global__ __launch_bounds__(256)
void nv_copy_tail(const float4* __restrict__ src, float4* __restrict__ dst,
                  int start, int n4)
{
    int i = start + blockIdx.x * 256 + threadIdx.x;
    if (i < n4) dst[i] = src[i];
}

// ---------------------------------------------------------------------------
// Kernel 2: per-point trilinear gather + interpolate + sigmoid-weighted
// scatter-add into the cloned grid (hardware f32 global atomics; the voxel
// table is ~L2-resident after the clone, so atomics/gathers stay on-chip).
// ---------------------------------------------------------------------------
__global__ __launch_bounds__(256)
void nv_update(const int*   __restrict__ t,
               const float* __restrict__ pos,
               const float* __restrict__ lr,
               const float* __restrict__ sigma,
               const float* __restrict__ target,
               const float* __restrict__ vin,
               float*       __restrict__ value_out,
               float*       __restrict__ vout,
               int N)
{
    int i = blockIdx.x * 256 + threadIdx.x;
    if (i >= N) return;

    // min_idx = floor(pos_s/STEP); frac = mod(pos_s, STEP)/STEP
    float psx = pos[3 * i + 0] - NV_RMIN;
    float psy = pos[3 * i + 1] - NV_RMIN;
    float psz = pos[3 * i + 2] - NV_RMIN;

    float fx = floorf(psx / NV_STEP);
    float fy = floorf(psy / NV_STEP);
    float fz = floorf(psz / NV_STEP);
    int ix = (int)fx, iy = (int)fy, iz = (int)fz;

    float u = (psx - fx * NV_STEP) / NV_STEP;
    float v = (psy - fy * NV_STEP) / NV_STEP;
    float w = (psz - fz * NV_STEP) / NV_STEP;

    int xs[2] = { min(max(ix,     0), NV_G - 1), min(max(ix + 1, 0), NV_G - 1) };
    int ys[2] = { min(max(iy,     0), NV_G - 1), min(max(iy + 1, 0), NV_G - 1) };
    int zs[2] = { min(max(iz,     0), NV_G - 1), min(max(iz + 1, 0), NV_G - 1) };
    float cu[2] = { 1.0f - u, u };
    float cv[2] = { 1.0f - v, v };
    float cw[2] = { 1.0f - w, w };

    const int tbase = t[i] * (NV_G * NV_G * NV_G * 3);

    float sw = 1.0f - __expf(-sigma[i]);
    float a  = lr[0] * sw;

    float tx = target[3 * i + 0];
    float ty = target[3 * i + 1];
    float tz = target[3 * i + 2];

    float accx = 0.0f, accy = 0.0f, accz = 0.0f;

    // Offsets table order: dx = k&1, dy = (k>>1)&1, dz = (k>>2)&1
#pragma unroll
    for (int k = 0; k < 8; ++k) {
        int dx = k & 1, dy = (k >> 1) & 1, dz = (k >> 2) & 1;
        float c = cu[dx] * cv[dy] * cw[dz];
        int off = tbase + ((xs[dx] * NV_G + ys[dy]) * NV_G + zs[dz]) * 3;

        float vx = vin[off + 0];
        float vy = vin[off + 1];
        float vz = vin[off + 2];

        accx += c * vx;
        accy += c * vy;
        accz += c * vz;

        // update_lambda = sigmoid(lr * sigma_weight * coef)
        float lam = 1.0f / (1.0f + __expf(-(a * c)));

        atomicAdd(&vout[off + 0], lam * (tx - vx));
        atomicAdd(&vout[off + 1], lam * (ty - vy));
        atomicAdd(&vout[off + 2], lam * (tz - vz));
    }

    value_out[3 * i + 0] = accx;
    value_out[3 * i + 1] = accy;
    value_out[3 * i + 2] = accz;
}

// ---------------------------------------------------------------------------
// Host launcher
// Inputs (setup_inputs order): t(i32,N), pos(f32,N*3), lr(f32,1),
// sigma(f32,N), target_norm(f32,N*3), voxel_array(f32, 8*128^3*3).
// Output: [value (N*3) | new_voxel (8*128^3*3)] f32, concatenated.
// ---------------------------------------------------------------------------
extern "C" void kernel_launch(void* const* d_in, const int* in_sizes, int n_in,
                              void* d_out, int out_size, void* d_ws, size_t ws_size,
                              hipStream_t stream)
{
    (void)n_in; (void)out_size; (void)d_ws; (void)ws_size;

    const int*   t      = (const int*)  d_in[0];
    const float* pos    = (const float*)d_in[1];
    const float* lr     = (const float*)d_in[2];
    const float* sigma  = (const float*)d_in[3];
    const float* target = (const float*)d_in[4];
    const float* voxel  = (const float*)d_in[5];

    const int N = in_sizes[0];          // 1048576
    const int V = in_sizes[5];          // 8*128^3*3 = 50331648 floats

    float* value_out = (float*)d_out;
    float* vout      = value_out + (size_t)N * 3;

    // 1) Clone voxel grid: full tiles via async global<->LDS DMA path,
    //    remainder (none at reference sizes) via scalar tail kernel.
    const int n4    = V / 4;            // float4 count (V divisible by 4)
    const int tile  = 256 * NV_CH;      // float4s per block (4096)
    const int nfull = n4 / tile;        // 3072 full tiles at reference sizes
    const int tail  = n4 - nfull * tile;

    if (nfull > 0) {
        nv_copy_async<<<nfull, 256, 0, stream>>>((const float4*)voxel,
                                                 (float4*)vout);
    }
    if (tail > 0) {
        nv_copy_tail<<<(tail + 255) / 256, 256, 0, stream>>>(
            (const float4*)voxel, (float4*)vout, nfull * tile, n4);
    }

    // 2) Gather/interpolate + atomic scatter-add of the update.
    nv_update<<<(N + 255) / 256, 256, 0, stream>>>(
        t, pos, lr, sigma, target, voxel, value_out, vout, N);
}